// ImprovedGCNModel_63488206570124
// MI455X (gfx1250) — compile-verified
//
#include <hip/hip_runtime.h>
#include <hip/hip_bf16.h>

// ---------------------------------------------------------------------------
// GCN model for MI455X (gfx1250, wave32, WMMA).
// Dense GEMMs use V_WMMA_F32_16X16X4_F32 (fp32-faithful, matrix pipe).
// Sparse aggregation uses fp32 global atomics (L2-resident: 51MB node table
// fits in the 192MB L2).
// ---------------------------------------------------------------------------

#define NN 100000
#define EE 1600000
#define HH 128

typedef __attribute__((ext_vector_type(2))) float v2f;
typedef __attribute__((ext_vector_type(8))) float v8f;

// ---- WMMA 16x16 output tile, K=128, fp32 -----------------------------------
// A (16x4 f32): lanes 0-15 hold M=lane, K=k0+{0,1}; lanes 16-31 hold M=lane-16,
// K=k0+{2,3}.  B (4x16): lanes 0-15 hold N=lane, K=k0+{0,1} in v0/v1; lanes
// 16-31 hold K=k0+{2,3}.  C/D (16x16 f32): VGPR r = row r (lanes 0-15) /
// row r+8 (lanes 16-31), col = lane&15.
__device__ __forceinline__ v8f wmma_tile_k128(const float* __restrict__ X,
                                              const float* __restrict__ W,
                                              int rowBase, int colBase, int lane) {
  v8f acc = {};
  const int half = lane >> 4;
  const int m    = lane & 15;
  const float* __restrict__ Ar = X + (size_t)(rowBase + m) * HH;
  const float* __restrict__ Wc = W + colBase + m;
#pragma unroll
  for (int k0 = 0; k0 < HH; k0 += 4) {
    const int ka = k0 + 2 * half;
    v2f a, b;
    a.x = Ar[ka];
    a.y = Ar[ka + 1];
    b.x = Wc[(size_t)ka * HH];
    b.y = Wc[(size_t)(ka + 1) * HH];
    acc = __builtin_amdgcn_wmma_f32_16x16x4_f32(false, a, false, b,
                                                (short)0, acc, false, false);
  }
  return acc;
}

// ---- plain GEMM: Y[N,128] = X[N,128] @ W[128,128] --------------------------
__global__ void gemm128_kernel(const float* __restrict__ X,
                               const float* __restrict__ W,
                               float* __restrict__ Y) {
  const int lane    = threadIdx.x & 31;
  const int wave    = threadIdx.x >> 5;
  const int rowBase = blockIdx.x << 4;
  const int colBase = wave << 4;
  v8f acc = wmma_tile_k128(X, W, rowBase, colBase, lane);
  const int half = lane >> 4;
  const int m    = lane & 15;
#pragma unroll
  for (int r = 0; r < 8; ++r)
    Y[(size_t)(rowBase + r + 8 * half) * HH + colBase + m] = acc[r];
}

// ---- GEMM + bias + BatchNorm(eval) + ReLU epilogue -------------------------
__global__ void gemm_pre_kernel(const float* __restrict__ X,
                                const float* __restrict__ W,
                                const float* __restrict__ bias,
                                const float* __restrict__ gamma,
                                const float* __restrict__ beta,
                                const float* __restrict__ mean,
                                const float* __restrict__ var,
                                float* __restrict__ Y) {
  const int lane    = threadIdx.x & 31;
  const int wave    = threadIdx.x >> 5;
  const int rowBase = blockIdx.x << 4;
  const int colBase = wave << 4;
  v8f acc = wmma_tile_k128(X, W, rowBase, colBase, lane);
  const int half = lane >> 4;
  const int m    = lane & 15;
  const int col  = colBase + m;
  const float bi = bias[col];
  const float sc = rsqrtf(var[col] + 1e-5f) * gamma[col];
  const float mb = mean[col];
  const float be = beta[col];
#pragma unroll
  for (int r = 0; r < 8; ++r) {
    float v = (acc[r] + bi - mb) * sc + be;
    v = fmaxf(v, 0.0f);
    Y[(size_t)(rowBase + r + 8 * half) * HH + colBase + m] = v;
  }
}

// ---- degree / dinv ---------------------------------------------------------
__global__ void fill_kernel(float* __restrict__ p, float v, int n) {
  int i = blockIdx.x * blockDim.x + threadIdx.x;
  if (i < n) p[i] = v;
}

__global__ void deg_kernel(const int* __restrict__ ei,
                           const float* __restrict__ ew,
                           float* __restrict__ deg) {
  int e = blockIdx.x * blockDim.x + threadIdx.x;
  if (e >= EE) return;
  int d = ei[EE + e];
  __hip_atomic_fetch_add(&deg[d], ew[e], __ATOMIC_RELAXED,
                         __HIP_MEMORY_SCOPE_AGENT);
}

__global__ void dinv_kernel(float* __restrict__ dinv) {
  int v = blockIdx.x * blockDim.x + threadIdx.x;
  if (v >= NN) return;
  dinv[v] = rsqrtf(dinv[v] + 1.0f);  // +1 for self-loop; always > 0
}

// ---- C[v] = dinv[v]^2 * B[v]  (self-loop term initializes accumulator) -----
__global__ void selfinit_kernel(const float* __restrict__ B,
                                const float* __restrict__ dinv,
                                float* __restrict__ C) {
  int t = blockIdx.x * blockDim.x + threadIdx.x;   // NN*32 float4 chunks
  int v = t >> 5;
  int q = t & 31;
  float dv = dinv[v];
  float c  = dv * dv;
  float4 val = ((const float4*)(B + (size_t)v * HH))[q];
  val.x *= c; val.y *= c; val.z *= c; val.w *= c;
  ((float4*)(C + (size_t)v * HH))[q] = val;
}

// ---- edge scatter: one wave per edge, 4 floats per lane --------------------
__global__ void edge_scatter_kernel(const int* __restrict__ ei,
                                    const float* __restrict__ ew,
                                    const float* __restrict__ dinv,
                                    const float* __restrict__ T,
                                    float* __restrict__ C) {
  int gid  = blockIdx.x * blockDim.x + threadIdx.x;
  int e    = gid >> 5;
  int lane = gid & 31;
  if (e >= EE) return;
  int s = ei[e];
  int d = ei[EE + e];
  float coef = dinv[s] * ew[e] * dinv[d];
  float4 v = ((const float4*)(T + (size_t)s * HH))[lane];
  float* o = C + (size_t)d * HH + lane * 4;
  __hip_atomic_fetch_add(o + 0, coef * v.x, __ATOMIC_RELAXED, __HIP_MEMORY_SCOPE_AGENT);
  __hip_atomic_fetch_add(o + 1, coef * v.y, __ATOMIC_RELAXED, __HIP_MEMORY_SCOPE_AGENT);
  __hip_atomic_fetch_add(o + 2, coef * v.z, __ATOMIC_RELAXED, __HIP_MEMORY_SCOPE_AGENT);
  __hip_atomic_fetch_add(o + 3, coef * v.w, __ATOMIC_RELAXED, __HIP_MEMORY_SCOPE_AGENT);
}

// ---- C = relu(C + bias) ----------------------------------------------------
__global__ void bias_relu_kernel(float* __restrict__ C,
                                 const float* __restrict__ bias) {
  int t = blockIdx.x * blockDim.x + threadIdx.x;   // NN*32 float4 chunks
  int v = t >> 5;
  int q = t & 31;
  int f = q * 4;
  float4 val = ((float4*)(C + (size_t)v * HH))[q];
  float4 b   = *(const float4*)(bias + f);
  val.x = fmaxf(val.x + b.x, 0.0f);
  val.y = fmaxf(val.y + b.y, 0.0f);
  val.z = fmaxf(val.z + b.z, 0.0f);
  val.w = fmaxf(val.w + b.w, 0.0f);
  ((float4*)(C + (size_t)v * HH))[q] = val;
}

// ---- C = C + bias + residual A ---------------------------------------------
__global__ void bias_residual_kernel(float* __restrict__ C,
                                     const float* __restrict__ bias,
                                     const float* __restrict__ A) {
  int t = blockIdx.x * blockDim.x + threadIdx.x;
  int v = t >> 5;
  int q = t & 31;
  int f = q * 4;
  float4 val = ((float4*)(C + (size_t)v * HH))[q];
  float4 b   = *(const float4*)(bias + f);
  float4 rr  = ((const float4*)(A + (size_t)v * HH))[q];
  val.x += b.x + rr.x;
  val.y += b.y + rr.y;
  val.z += b.z + rr.z;
  val.w += b.w + rr.w;
  ((float4*)(C + (size_t)v * HH))[q] = val;
}

// ---- LayerNorm(+bias)+ReLU in place: one wave32 per row --------------------
__global__ void ln_relu_kernel(float* __restrict__ Z,
                               const float* __restrict__ bias,
                               const float* __restrict__ g,
                               const float* __restrict__ b) {
  int lane = threadIdx.x & 31;
  int wave = threadIdx.x >> 5;
  int row  = blockIdx.x * 8 + wave;
  if (row >= NN) return;
  float* z = Z + (size_t)row * HH;
  int f = lane * 4;
  float4 v  = *(float4*)(z + f);
  float4 bb = *(const float4*)(bias + f);
  v.x += bb.x; v.y += bb.y; v.z += bb.z; v.w += bb.w;
  float s  = v.x + v.y + v.z + v.w;
  float sq = v.x * v.x + v.y * v.y + v.z * v.z + v.w * v.w;
#pragma unroll
  for (int m = 16; m > 0; m >>= 1) {
    s  += __shfl_xor(s, m, 32);
    sq += __shfl_xor(sq, m, 32);
  }
  float mu  = s * (1.0f / HH);
  float var = sq * (1.0f / HH) - mu * mu;
  float inv = rsqrtf(var + 1e-5f);
  float4 gg  = *(const float4*)(g + f);
  float4 bb2 = *(const float4*)(b + f);
  v.x = fmaxf((v.x - mu) * inv * gg.x + bb2.x, 0.0f);
  v.y = fmaxf((v.y - mu) * inv * gg.y + bb2.y, 0.0f);
  v.z = fmaxf((v.z - mu) * inv * gg.z + bb2.z, 0.0f);
  v.w = fmaxf((v.w - mu) * inv * gg.w + bb2.w, 0.0f);
  *(float4*)(z + f) = v;
}

// ---- head: out[N,3] = Z[N,128] @ Wc2[128,3] + bc2 --------------------------
__global__ void head_kernel(const float* __restrict__ Z,
                            const float* __restrict__ Wc2,
                            const float* __restrict__ bc2,
                            float* __restrict__ out) {
  int row = blockIdx.x * blockDim.x + threadIdx.x;
  if (row >= NN) return;
  float a0 = bc2[0], a1 = bc2[1], a2 = bc2[2];
  const float* z = Z + (size_t)row * HH;
#pragma unroll 8
  for (int k = 0; k < HH; ++k) {
    float v = z[k];
    a0 += v * Wc2[k * 3 + 0];
    a1 += v * Wc2[k * 3 + 1];
    a2 += v * Wc2[k * 3 + 2];
  }
  out[row * 3 + 0] = a0;
  out[row * 3 + 1] = a1;
  out[row * 3 + 2] = a2;
}

// ---------------------------------------------------------------------------
extern "C" void kernel_launch(void* const* d_in, const int* in_sizes, int n_in,
                              void* d_out, int out_size, void* d_ws, size_t ws_size,
                              hipStream_t stream) {
  (void)in_sizes; (void)n_in; (void)out_size; (void)ws_size;
  const float* x     = (const float*)d_in[0];
  const int*   ei    = (const int*)  d_in[1];
  const float* ea    = (const float*)d_in[2];
  const float* W_pre = (const float*)d_in[3];
  const float* b_pre = (const float*)d_in[4];
  const float* bn_g  = (const float*)d_in[5];
  const float* bn_b  = (const float*)d_in[6];
  const float* bn_m  = (const float*)d_in[7];
  const float* bn_v  = (const float*)d_in[8];
  const float* W1    = (const float*)d_in[9];
  const float* b1    = (const float*)d_in[10];
  const float* W2    = (const float*)d_in[11];
  const float* b2    = (const float*)d_in[12];
  const float* Wc1   = (const float*)d_in[13];
  const float* bc1   = (const float*)d_in[14];
  const float* ln_g  = (const float*)d_in[15];
  const float* ln_b  = (const float*)d_in[16];
  const float* Wc2   = (const float*)d_in[17];
  const float* bc2   = (const float*)d_in[18];
  float* out = (float*)d_out;

  // workspace: dinv[N] | A=h0/residual[N*128] | B=gemm temp[N*128] | C=agg[N*128]
  float* dinv = (float*)d_ws;
  float* A = dinv + NN;
  float* B = A + (size_t)NN * HH;
  float* C = B + (size_t)NN * HH;

  const int ELEM_BLOCKS = (NN * 32) / 256;       // 12500 (float4 elementwise)
  const int GEMM_BLOCKS = NN / 16;               // 6250
  const int EDGE_BLOCKS = (EE * 32) / 256;       // 200000 (wave per edge)

  // normalization coefficients
  fill_kernel<<<(NN + 255) / 256, 256, 0, stream>>>(dinv, 0.0f, NN);
  deg_kernel<<<(EE + 255) / 256, 256, 0, stream>>>(ei, ea, dinv);
  dinv_kernel<<<(NN + 255) / 256, 256, 0, stream>>>(dinv);

  // input preprocess: Linear -> BN(eval) -> ReLU   => A (= residual)
  gemm_pre_kernel<<<GEMM_BLOCKS, 256, 0, stream>>>(x, W_pre, b_pre, bn_g, bn_b,
                                                   bn_m, bn_v, A);

  // GCN conv 1: t = A@W1; agg; relu(+b1)           => C = h1
  gemm128_kernel<<<GEMM_BLOCKS, 256, 0, stream>>>(A, W1, B);
  selfinit_kernel<<<ELEM_BLOCKS, 256, 0, stream>>>(B, dinv, C);
  edge_scatter_kernel<<<EDGE_BLOCKS, 256, 0, stream>>>(ei, ea, dinv, B, C);
  bias_relu_kernel<<<ELEM_BLOCKS, 256, 0, stream>>>(C, b1);

  // GCN conv 2: t = C@W2; agg; +b2 +residual       => C = h
  gemm128_kernel<<<GEMM_BLOCKS, 256, 0, stream>>>(C, W2, B);
  selfinit_kernel<<<ELEM_BLOCKS, 256, 0, stream>>>(B, dinv, C);
  edge_scatter_kernel<<<EDGE_BLOCKS, 256, 0, stream>>>(ei, ea, dinv, B, C);
  bias_residual_kernel<<<ELEM_BLOCKS, 256, 0, stream>>>(C, b2, A);

  // classifier: z = C@Wc1 + bc1 -> LN -> ReLU       => B = z
  gemm128_kernel<<<GEMM_BLOCKS, 256, 0, stream>>>(C, Wc1, B);
  ln_relu_kernel<<<NN / 8, 256, 0, stream>>>(B, bc1, ln_g, ln_b);

  // head: out = B@Wc2 + bc2
  head_kernel<<<(NN + 255) / 256, 256, 0, stream>>>(B, Wc2, bc2, out);
}